// make_blocks_38860864094557
// MI455X (gfx1250) — compile-verified
//
#include <hip/hip_runtime.h>

// Problem constants (from reference): B=64, L=2048, D=60, P=16, PS=30
#define NB   64
#define NL   2048
#define ND   60
#define NP   16
#define NPS  30
#define NCELL (NPS * NPS)          // 900 cells per patch
#define NK    (2 * ND + 1)         // 121 channels per cell
#define PATCH_FLOATS (NCELL * NK)  // 108900 floats per patch
#define PATCH_V4     (PATCH_FLOATS / 4)  // 27225 float4 stores (exact)
#define ROW_F   (NPS * ND)         // 1800 floats per rows/cols segment
#define ROW_V4  (ROW_F / 4)        // 450 float4
#define GEO_V4  (NCELL / 4)        // 225 float4
// Fused LDS layout: [0,1800) rows | [1800,3600) cols | [3600,4500) geo
#define LDS_COLS 1800u
#define LDS_GEO  3600u
#define LDS_F    4500

typedef float __attribute__((ext_vector_type(4))) f4;
typedef int   __attribute__((ext_vector_type(4))) v4i;

#if __has_builtin(__builtin_amdgcn_global_load_async_to_lds_b128)
#define HAVE_ASYNC_LDS 1
typedef __attribute__((address_space(1))) v4i* gvp;  // global (AS1) int4*
typedef __attribute__((address_space(3))) v4i* lvp;  // LDS (AS3) int4*
#else
#define HAVE_ASYNC_LDS 0
#endif

__device__ __forceinline__ void wait_asynccnt_zero() {
#if __has_builtin(__builtin_amdgcn_s_wait_asynccnt)
    __builtin_amdgcn_s_wait_asynccnt(0);
#else
    asm volatile("s_wait_asynccnt 0" ::: "memory");
#endif
}

__global__ __launch_bounds__(256) void make_blocks_38860864094557_kernel(
    const float* __restrict__ seq1, const float* __restrict__ seq2,
    const float* __restrict__ geo, const int* __restrict__ patches,
    float* __restrict__ out)
{
    // 18 KB LDS staging per block (320 KB/WGP -> high occupancy).
    __shared__ alignas(16) float lds[LDS_F];

    const int bp  = blockIdx.x;          // patch id in [0, B*P)
    const int b   = bp >> 4;             // bp / 16
    const int tid = threadIdx.x;

    const int r0 = patches[bp * 2 + 0];
    const int c0 = patches[bp * 2 + 1];

    // All three source regions are fully contiguous and 16B aligned
    // (offsets are multiples of D*4 = 240B or 900*4 = 3600B).
    const float* rbase = seq1 + ((size_t)b * NL + r0) * ND;
    const float* cbase = seq2 + ((size_t)b * NL + c0) * ND;
    const float* gbase = geo + (size_t)bp * NCELL;

#if HAVE_ASYNC_LDS
    {
        // Step 1: reinterpret (and drop const) in the generic address space.
        v4i* rgen = (v4i*)rbase;
        v4i* cgen = (v4i*)cbase;
        v4i* ggen = (v4i*)gbase;
        v4i* ldsgen = (v4i*)lds;
        // Step 2: pure address-space casts (generic -> AS1 / AS3).
        gvp rg = (gvp)rgen;
        gvp cg = (gvp)cgen;
        gvp gg = (gvp)ggen;
        lvp ll = (lvp)ldsgen;

        for (int i = tid; i < ROW_V4; i += 256)
            __builtin_amdgcn_global_load_async_to_lds_b128(rg + i, ll + i, 0, 0);
        for (int i = tid; i < ROW_V4; i += 256)
            __builtin_amdgcn_global_load_async_to_lds_b128(cg + i, ll + (LDS_COLS / 4) + i, 0, 0);
        for (int i = tid; i < GEO_V4; i += 256)
            __builtin_amdgcn_global_load_async_to_lds_b128(gg + i, ll + (LDS_GEO / 4) + i, 0, 0);
        wait_asynccnt_zero();   // each wave's async LDS writes have landed
    }
#else
    {
        const f4* rsrc = (const f4*)rbase;
        const f4* csrc = (const f4*)cbase;
        const f4* gsrc = (const f4*)gbase;
        f4* ldsv = (f4*)lds;
        for (int i = tid; i < ROW_V4; i += 256) ldsv[i] = rsrc[i];
        for (int i = tid; i < ROW_V4; i += 256) ldsv[(LDS_COLS / 4) + i] = csrc[i];
        for (int i = tid; i < GEO_V4; i += 256) ldsv[(LDS_GEO / 4) + i] = gsrc[i];
    }
#endif
    __syncthreads();  // make LDS visible across all 8 waves

    // Flat, fully-coalesced streaming store of the 435.6 KB output patch.
    // Flat element t = cell*121 + k, cell = a*30 + c:
    //   k <  60 -> lds[c*60 + k]                  (rows, broadcast over a)
    //   k < 120 -> lds[1800 + a*60 + (k-60)]      (cols, broadcast over c)
    //   k == 120-> lds[3600 + cell]               (geo)
    f4* __restrict__ outv = (f4*)(out + (size_t)bp * PATCH_FLOATS);

    for (int i = tid; i < PATCH_V4; i += 256) {
        const unsigned t = (unsigned)i * 4u;
        // Two divisions per float4 (magic-mul); the 4 elements advance
        // (cell,k,a,c) incrementally with compare/select only.
        unsigned cell = t / (unsigned)NK;
        unsigned k    = t - cell * (unsigned)NK;
        unsigned a    = cell / (unsigned)NPS;
        unsigned c    = cell - a * (unsigned)NPS;
        f4 v;
#pragma unroll
        for (int j = 0; j < 4; ++j) {
            unsigned idx;
            if (k < (unsigned)ND)
                idx = c * (unsigned)ND + k;
            else if (k < (unsigned)(2 * ND))
                idx = LDS_COLS + a * (unsigned)ND + (k - (unsigned)ND);
            else
                idx = LDS_GEO + cell;
            v[j] = lds[idx];             // single ds_load_b32 per element
            // advance one flat element (k can wrap at most once per step)
            ++k;
            if (k == (unsigned)NK) { k = 0u; ++cell; ++c; }
            if (c == (unsigned)NPS) { c = 0u; ++a; }
        }
        // 446 MB write stream > 192 MB L2: bypass with non-temporal b128.
        __builtin_nontemporal_store(v, outv + i);
    }
}

extern "C" void kernel_launch(void* const* d_in, const int* in_sizes, int n_in,
                              void* d_out, int out_size, void* d_ws, size_t ws_size,
                              hipStream_t stream) {
    const float* seq1    = (const float*)d_in[0];
    const float* seq2    = (const float*)d_in[1];
    const float* geo     = (const float*)d_in[2];
    const int*   patches = (const int*)d_in[3];
    float* out = (float*)d_out;

    // One 256-thread block (8 wave32) per (b,p) patch: 1024 blocks.
    make_blocks_38860864094557_kernel<<<NB * NP, 256, 0, stream>>>(
        seq1, seq2, geo, patches, out);
}